// Attention_19997367730322
// MI455X (gfx1250) — compile-verified
//
#include <hip/hip_runtime.h>

typedef __attribute__((ext_vector_type(2))) float v2f;
typedef __attribute__((ext_vector_type(8))) float v8f;

#define NUM_HEADS     32
#define HEAD_DIM      128
#define NUM_KV_HEADS  8
#define GQA           4
#define BLOCK_SZ      16
#define MAX_BLK       128
#define QK_SCALE      0.08838834764831845f
#define WAVES         8
#define BLOCK_THREADS (WAVES * 32)

__device__ __forceinline__ v8f wmma_f32(v2f a, v2f b, v8f c) {
  // D = A(16x4 f32) * B(4x16 f32) + C(16x16 f32), exact fp32 matrix op
  return __builtin_amdgcn_wmma_f32_16x16x4_f32(false, a, false, b, (short)0, c,
                                               false, false);
}

__global__ __launch_bounds__(BLOCK_THREADS) void paged_attn_kernel(
    const float* __restrict__ q, const float* __restrict__ k_cache,
    const float* __restrict__ v_cache, const int* __restrict__ block_tables,
    const int* __restrict__ context_lens, float* __restrict__ out) {
  const int bkv  = blockIdx.x;          // 0 .. B*KVH-1
  const int b    = bkv >> 3;
  const int kvh  = bkv & 7;
  const int tid  = threadIdx.x;
  const int w    = tid >> 5;            // wave id within WG
  const int lane = tid & 31;
  const int t    = lane & 15;           // column / row-within-half
  const int half = lane >> 4;           // 0: K=0,1   1: K=2,3 (WMMA f32 A/B layout)

  __shared__ __align__(16) float sS[WAVES][64];   // P transpose buffer (per wave)
  __shared__ float sO[WAVES][GQA * HEAD_DIM];     // per-wave partial O
  __shared__ float sM[WAVES][GQA];
  __shared__ float sL[WAVES][GQA];

  const int L      = context_lens[b];
  const int npages = (L + BLOCK_SZ - 1) >> 4;
  const size_t row_stride = (size_t)NUM_KV_HEADS * HEAD_DIM; // 1024 floats / token row

  // ---- Q A-fragments in registers: lane's row = t (valid heads: t<4, rest zero) ----
  v2f qa[32];
  if (t < GQA) {
    const float* qrow = q + ((size_t)b * NUM_HEADS + kvh * GQA + t) * HEAD_DIM;
#pragma unroll
    for (int c = 0; c < 32; ++c)
      qa[c] = *(const v2f*)(qrow + c * 4 + 2 * half);
  } else {
#pragma unroll
    for (int c = 0; c < 32; ++c) qa[c] = (v2f){0.0f, 0.0f};
  }

  v8f acc[8];
#pragma unroll
  for (int j = 0; j < 8; ++j) acc[j] = (v8f){0, 0, 0, 0, 0, 0, 0, 0};
  float m[GQA], l[GQA];
#pragma unroll
  for (int g = 0; g < GQA; ++g) { m[g] = -__builtin_inff(); l[g] = 0.0f; }

  // Hoisted block id for the first page this wave owns.
  int blk = 0;
  if (w < npages) {
    blk = block_tables[b * MAX_BLK + w];
    if (blk < 0) blk = 0;
  }

  for (int page = w; page < npages; page += WAVES) {
    // ---- hoist next page's block id + prefetch its K/V lines (hide indirect chain) ----
    const int nextpage = page + WAVES;
    int nblk = 0;
    if (nextpage < npages) {
      nblk = block_tables[b * MAX_BLK + nextpage];
      if (nblk < 0) nblk = 0;
      // lane (t, half) covers row t, bytes [half*256, half*256+256) in 2x128B lines
      const size_t pf_off =
          ((size_t)nblk * BLOCK_SZ + t) * row_stride + kvh * HEAD_DIM + half * 64;
      __builtin_prefetch(k_cache + pf_off, 0, 0);
      __builtin_prefetch(k_cache + pf_off + 32, 0, 0);
      __builtin_prefetch(v_cache + pf_off, 0, 0);
      __builtin_prefetch(v_cache + pf_off + 32, 0, 0);
    }

    // ---- S = Q * K^T : 32 chained f32 WMMAs over D=128 ----
    const float* kbase =
        k_cache + ((size_t)blk * BLOCK_SZ + t) * row_stride + kvh * HEAD_DIM;
    v8f s = (v8f){0, 0, 0, 0, 0, 0, 0, 0};
#pragma unroll
    for (int c = 0; c < 32; ++c) {
      v2f kb = *(const v2f*)(kbase + c * 4 + 2 * half);
      s = wmma_f32(qa[c], kb, s);
    }

    // lane (<16) holds S[g][t] for g=0..3 in s[0..3] (C-layout). Scale + mask tail.
    const bool valid = (page * BLOCK_SZ + t) < L;
    float sv[GQA], tm[GQA];
#pragma unroll
    for (int g = 0; g < GQA; ++g) {
      sv[g] = valid ? s[g] * QK_SCALE : -__builtin_inff();
      tm[g] = sv[g];
    }
    // row max across 16-token tile (xor masks <16 stay within each lane-half)
#pragma unroll
    for (int off = 8; off >= 1; off >>= 1)
#pragma unroll
      for (int g = 0; g < GQA; ++g)
        tm[g] = fmaxf(tm[g], __shfl_xor(tm[g], off, 32));

    float corr[GQA], p[GQA], tl[GQA];
#pragma unroll
    for (int g = 0; g < GQA; ++g) {
      float mn = fmaxf(m[g], tm[g]);
      corr[g] = __expf(m[g] - mn);   // exp(-inf - finite) = 0 on first tile
      m[g]    = mn;
      p[g]    = __expf(sv[g] - mn);  // masked cols -> 0
      tl[g]   = p[g];
    }
#pragma unroll
    for (int off = 8; off >= 1; off >>= 1)
#pragma unroll
      for (int g = 0; g < GQA; ++g) tl[g] += __shfl_xor(tl[g], off, 32);
#pragma unroll
    for (int g = 0; g < GQA; ++g) l[g] = l[g] * corr[g] + tl[g];

    // rescale accumulator rows (rows 4..7 are identically zero; leave them)
#pragma unroll
    for (int j = 0; j < 8; ++j)
#pragma unroll
      for (int g = 0; g < GQA; ++g) acc[j][g] *= corr[g];

    // ---- transpose P: C-layout (column per lane) -> A-layout (row per lane) via LDS ----
    asm volatile("" ::: "memory");
    if (lane < 16) {
#pragma unroll
      for (int g = 0; g < GQA; ++g) sS[w][g * 16 + t] = p[g];
    }
    asm volatile("s_wait_dscnt 0x0" ::: "memory");  // cross-lane store->load, same wave

    v2f pa[4];
#pragma unroll
    for (int c = 0; c < 4; ++c) {
      if (t < GQA)
        pa[c] = *(const v2f*)&sS[w][t * 16 + c * 4 + 2 * half];
      else
        pa[c] = (v2f){0.0f, 0.0f};
    }

    // ---- O += P * V : 4 chained K-slices x 8 d-tiles ----
    const float* vbase =
        v_cache + (size_t)blk * BLOCK_SZ * row_stride + kvh * HEAD_DIM;
#pragma unroll
    for (int j = 0; j < 8; ++j) {
      const float* vcol = vbase + j * 16 + t;  // column d = j*16 + n
#pragma unroll
      for (int c = 0; c < 4; ++c) {
        const int r = c * 4 + 2 * half;        // token rows r, r+1 for this lane
        v2f vb;
        vb.x = vcol[(size_t)r * row_stride];
        vb.y = vcol[(size_t)(r + 1) * row_stride];
        acc[j] = wmma_f32(pa[c], vb, acc[j]);
      }
    }

    blk = nblk;  // carry hoisted block id into next iteration
  }

  // ---- cross-wave combine via LDS (log-sum-exp merge of 8 partials) ----
  if (lane == 0) {
#pragma unroll
    for (int g = 0; g < GQA; ++g) { sM[w][g] = m[g]; sL[w][g] = l[g]; }
  }
  if (lane < 16) {
#pragma unroll
    for (int j = 0; j < 8; ++j)
#pragma unroll
      for (int g = 0; g < GQA; ++g)
        sO[w][g * HEAD_DIM + j * 16 + lane] = acc[j][g];
  }
  __syncthreads();

#pragma unroll
  for (int it = 0; it < (GQA * HEAD_DIM) / BLOCK_THREADS; ++it) {
    const int idx = tid + it * BLOCK_THREADS;
    const int g   = idx >> 7;
    const int d   = idx & 127;
    float gm = -__builtin_inff();
#pragma unroll
    for (int ww = 0; ww < WAVES; ++ww) gm = fmaxf(gm, sM[ww][g]);
    float num = 0.0f, den = 0.0f;
#pragma unroll
    for (int ww = 0; ww < WAVES; ++ww) {
      const float wt = __expf(sM[ww][g] - gm);
      den += wt * sL[ww][g];
      num += wt * sO[ww][g * HEAD_DIM + d];
    }
    out[((size_t)b * NUM_HEADS + kvh * GQA + g) * HEAD_DIM + d] = num / den;
  }
}

extern "C" void kernel_launch(void* const* d_in, const int* in_sizes, int n_in,
                              void* d_out, int out_size, void* d_ws,
                              size_t ws_size, hipStream_t stream) {
  const float* q            = (const float*)d_in[0];
  const float* k_cache      = (const float*)d_in[1];
  const float* v_cache      = (const float*)d_in[2];
  const int*   block_tables = (const int*)d_in[3];
  const int*   context_lens = (const int*)d_in[4];
  float*       out          = (float*)d_out;

  dim3 grid(64 * NUM_KV_HEADS);   // one WG per (batch, kv_head)
  dim3 block(BLOCK_THREADS);
  hipLaunchKernelGGL(paged_attn_kernel, grid, block, 0, stream, q, k_cache,
                     v_cache, block_tables, context_lens, out);
}